// Net_38079180047154
// MI455X (gfx1250) — compile-verified
//
#include <hip/hip_runtime.h>
#include <math.h>

typedef float v2f __attribute__((ext_vector_type(2)));
typedef float v8f __attribute__((ext_vector_type(8)));

#define NEG_SLOPE 0.2f
// monotonic float->uint key; MINKEY == key(-inf)
#define MINKEY 0x007FFFFFu

__device__ __forceinline__ unsigned f2key(float f) {
  unsigned b = __float_as_uint(f);
  return (b & 0x80000000u) ? ~b : (b | 0x80000000u);
}
__device__ __forceinline__ float key2f(unsigned k) {
  return (k & 0x80000000u) ? __uint_as_float(k & 0x7fffffffu) : __uint_as_float(~k);
}
__device__ __forceinline__ void edge_sd(const int* __restrict__ ei, int e, int Eraw,
                                        int& s, int& d) {
  if (e < Eraw) { s = ei[e]; d = ei[Eraw + e]; }
  else          { s = e - Eraw; d = e - Eraw; }   // self loop
}

// ---------------- init workspace ----------------
__global__ void init_ws(float* __restrict__ out1, float* __restrict__ out2,
                        unsigned* __restrict__ mk1, float* __restrict__ den1,
                        unsigned* __restrict__ mk2, float* __restrict__ den2, int n) {
  long long i = (long long)blockIdx.x * blockDim.x + threadIdx.x;
  if (i < (long long)n * 64) out1[i] = 0.f;
  if (i < (long long)n * 40) out2[i] = 0.f;
  if (i < (long long)n * 8)  { mk1[i] = MINKEY; den1[i] = 0.f; }
  if (i < (long long)n)      { mk2[i] = MINKEY; den2[i] = 0.f; }
}

// ---------------- fp32 WMMA GEMM: H = X[nrows,K] * W[K,ncols] ----------------
// One wave32 per 16-row stripe; NT 16x16 tiles wide (NT*16 >= ncols).
// Out-of-range rows/cols are CLAMPED (not predicated): D[:,n] depends only on
// B[:,n] and D[m,:] only on A[m,:], so clamped lanes corrupt only padded
// outputs that are never stored. Keeps EXEC all-1s through the k-loop (no
// divergent branches around the v_wmma chain) and lets A load as b64.
//
// A 16x4 layout: lanes 0-15 -> M, lane-half selects K pair {0,1}/{2,3}.
// B 4x16 layout: lanes 0-15 -> N, VGPR r holds K=r (lo half) / K=r+2 (hi half).
// D 16x16 layout: VGPR r -> M = r + 8*(lane>=16), N = lane&15.
template <int NT>
__global__ void gemm_wmma_f32(const float* __restrict__ X, const float* __restrict__ W,
                              float* __restrict__ Hout, int nrows, int K, int ncols) {
  int wave = (int)((blockIdx.x * blockDim.x + threadIdx.x) >> 5);
  int lane = threadIdx.x & 31;
  int ntm  = (nrows + 15) >> 4;
  if (wave >= ntm) return;

  int l15  = lane & 15;
  int half = lane >> 4;

  int row  = wave * 16 + l15;
  int rowc = row < nrows ? row : (nrows - 1);          // clamp (padded rows unused)

  int col[NT], colc[NT];
#pragma unroll
  for (int j = 0; j < NT; ++j) {
    col[j]  = j * 16 + l15;
    colc[j] = col[j] < ncols ? col[j] : (ncols - 1);   // clamp (padded cols unused)
  }

  const float* __restrict__ arow = X + (long long)rowc * K + 2 * half;
  const float* __restrict__ wrow = W + (long long)(2 * half) * ncols;

  v8f acc[NT];
#pragma unroll
  for (int j = 0; j < NT; ++j) acc[j] = (v8f){};

  for (int k0 = 0; k0 < K; k0 += 4) {
    v2f a = *(const v2f*)(arow + k0);                  // A[row][ka], A[row][ka+1]
    const float* wk = wrow + (long long)k0 * ncols;
    v2f b[NT];
#pragma unroll
    for (int j = 0; j < NT; ++j) {
      b[j][0] = wk[colc[j]];                           // B[ka  ][col]
      b[j][1] = wk[(long long)ncols + colc[j]];        // B[ka+1][col]
    }
#pragma unroll
    for (int j = 0; j < NT; ++j)
      acc[j] = __builtin_amdgcn_wmma_f32_16x16x4_f32(false, a, false, b[j],
                                                     (short)0, acc[j], false, false);
  }

  int mbase = wave * 16 + half * 8;
#pragma unroll
  for (int j = 0; j < NT; ++j) {
    if (col[j] >= ncols) continue;
#pragma unroll
    for (int r = 0; r < 8; ++r) {
      int m = mbase + r;
      if (m < nrows) Hout[(long long)m * ncols + col[j]] = acc[j][r];
    }
  }
}

// ---------------- per-(node,head) attention coefficients ----------------
__global__ void attn_coef(const float* __restrict__ Hm, const float* __restrict__ a_src,
                          const float* __restrict__ a_dst, float* __restrict__ as,
                          float* __restrict__ ad, int n, int H, int F) {
  int i = blockIdx.x * blockDim.x + threadIdx.x;
  if (i >= n * H) return;
  int node = i / H, h = i % H;
  const float* hp = Hm + (long long)node * H * F + (long long)h * F;
  float s = 0.f, d = 0.f;
  for (int f = 0; f < F; ++f) {
    float v = hp[f];
    s += v * a_src[h * F + f];
    d += v * a_dst[h * F + f];
  }
  as[i] = s; ad[i] = d;
}

// ---------------- edge pass 1: segment max of leaky-relu scores ----------------
__global__ void edge_max_k(const int* __restrict__ ei, const float* __restrict__ as,
                           const float* __restrict__ ad, unsigned* __restrict__ mk,
                           int Eraw, int Etot, int H) {
  int e = blockIdx.x * blockDim.x + threadIdx.x;
  if (e >= Etot) return;
  int s, d; edge_sd(ei, e, Eraw, s, d);
  for (int h = 0; h < H; ++h) {
    float v = as[s * H + h] + ad[d * H + h];
    v = v > 0.f ? v : NEG_SLOPE * v;
    atomicMax(&mk[d * H + h], f2key(v));
  }
}

// ---------------- edge pass 2: segment sum of exp(e - max) ----------------
__global__ void edge_denom_k(const int* __restrict__ ei, const float* __restrict__ as,
                             const float* __restrict__ ad, const unsigned* __restrict__ mk,
                             float* __restrict__ den, int Eraw, int Etot, int H) {
  int e = blockIdx.x * blockDim.x + threadIdx.x;
  if (e >= Etot) return;
  int s, d; edge_sd(ei, e, Eraw, s, d);
  for (int h = 0; h < H; ++h) {
    float v = as[s * H + h] + ad[d * H + h];
    v = v > 0.f ? v : NEG_SLOPE * v;
    atomicAdd(&den[d * H + h], expf(v - key2f(mk[d * H + h])));
  }
}

// ---------------- edge pass 3: out[dst] += h[src] * alpha ----------------
__global__ void edge_aggr_k(const int* __restrict__ ei, const float* __restrict__ as,
                            const float* __restrict__ ad, const unsigned* __restrict__ mk,
                            const float* __restrict__ den, const float* __restrict__ Hm,
                            float* __restrict__ out, int Eraw, int Etot, int H, int F) {
  int idx = blockIdx.x * blockDim.x + threadIdx.x;
  if (idx >= Etot * H) return;
  int e = idx / H, h = idx % H;
  int s, d; edge_sd(ei, e, Eraw, s, d);
  float v = as[s * H + h] + ad[d * H + h];
  v = v > 0.f ? v : NEG_SLOPE * v;
  float alpha = expf(v - key2f(mk[d * H + h])) / (den[d * H + h] + 1e-16f);
  const float* hp = Hm + (long long)s * H * F + (long long)h * F;
  float* op = out + (long long)d * H * F + (long long)h * F;
  for (int f = 0; f < F; ++f) atomicAdd(&op[f], hp[f] * alpha);
}

// ---------------- bias + ELU (in place) ----------------
__global__ void bias_elu_k(float* __restrict__ io, const float* __restrict__ bias,
                           long long total, int C) {
  long long i = (long long)blockIdx.x * blockDim.x + threadIdx.x;
  if (i >= total) return;
  float v = io[i] + bias[i % C];
  io[i] = v > 0.f ? v : (expf(v) - 1.f);
}

// ---------------- bias + log_softmax over 40 classes ----------------
__global__ void final_lsm_k(const float* __restrict__ out2, const float* __restrict__ bias,
                            float* __restrict__ dout, int n) {
  int i = blockIdx.x * blockDim.x + threadIdx.x;
  if (i >= n) return;
  float v[40];
  float mx = -INFINITY;
  for (int c = 0; c < 40; ++c) {
    v[c] = out2[(long long)i * 40 + c] + bias[c];
    mx = fmaxf(mx, v[c]);
  }
  float sum = 0.f;
  for (int c = 0; c < 40; ++c) sum += expf(v[c] - mx);
  float ls = logf(sum);
  for (int c = 0; c < 40; ++c) dout[(long long)i * 40 + c] = v[c] - mx - ls;
}

extern "C" void kernel_launch(void* const* d_in, const int* in_sizes, int n_in,
                              void* d_out, int out_size, void* d_ws, size_t ws_size,
                              hipStream_t stream) {
  const float* x    = (const float*)d_in[0];
  const int*   ei   = (const int*)  d_in[1];
  const float* W1   = (const float*)d_in[2];
  const float* aS1  = (const float*)d_in[3];
  const float* aD1  = (const float*)d_in[4];
  const float* b1   = (const float*)d_in[5];
  const float* W2   = (const float*)d_in[6];
  const float* aS2  = (const float*)d_in[7];
  const float* aD2  = (const float*)d_in[8];
  const float* b2   = (const float*)d_in[9];

  const int N    = in_sizes[0] / 256;   // 100000
  const int Eraw = in_sizes[1] / 2;     // 1600000
  const int Etot = Eraw + N;            // + self loops

  // carve workspace
  char* ws = (char*)d_ws;
  size_t off = 0;
  auto carve = [&](size_t bytes) -> void* {
    void* p = ws + off;
    off = (off + bytes + 255) & ~(size_t)255;
    return p;
  };
  float*    h1    = (float*)   carve((size_t)N * 64 * 4);
  float*    out1  = (float*)   carve((size_t)N * 64 * 4); // becomes hact after ELU
  float*    as1   = (float*)   carve((size_t)N * 8 * 4);
  float*    ad1   = (float*)   carve((size_t)N * 8 * 4);
  unsigned* mk1   = (unsigned*)carve((size_t)N * 8 * 4);
  float*    den1  = (float*)   carve((size_t)N * 8 * 4);
  float*    h2    = (float*)   carve((size_t)N * 40 * 4);
  float*    out2  = (float*)   carve((size_t)N * 40 * 4);
  float*    as2   = (float*)   carve((size_t)N * 4);
  float*    ad2   = (float*)   carve((size_t)N * 4);
  unsigned* mk2   = (unsigned*)carve((size_t)N * 4);
  float*    den2  = (float*)   carve((size_t)N * 4);
  (void)ws_size;

  const int B = 256;
  auto grid1 = [&](long long t) { return dim3((unsigned)((t + B - 1) / B)); };

  // init accumulators / max-keys
  init_ws<<<grid1((long long)N * 64), B, 0, stream>>>(out1, out2, mk1, den1, mk2, den2, N);

  const long long waves = (long long)((N + 15) / 16);  // one wave per 16-row stripe

  // ---- layer 1 ----
  gemm_wmma_f32<4><<<grid1(waves * 32), B, 0, stream>>>(x, W1, h1, N, 256, 64);
  attn_coef<<<grid1((long long)N * 8), B, 0, stream>>>(h1, aS1, aD1, as1, ad1, N, 8, 8);
  edge_max_k  <<<grid1(Etot), B, 0, stream>>>(ei, as1, ad1, mk1, Eraw, Etot, 8);
  edge_denom_k<<<grid1(Etot), B, 0, stream>>>(ei, as1, ad1, mk1, den1, Eraw, Etot, 8);
  edge_aggr_k <<<grid1((long long)Etot * 8), B, 0, stream>>>(ei, as1, ad1, mk1, den1, h1,
                                                             out1, Eraw, Etot, 8, 8);
  bias_elu_k<<<grid1((long long)N * 64), B, 0, stream>>>(out1, b1, (long long)N * 64, 64);

  // ---- layer 2 ----
  gemm_wmma_f32<3><<<grid1(waves * 32), B, 0, stream>>>(out1, W2, h2, N, 64, 40);
  attn_coef<<<grid1((long long)N), B, 0, stream>>>(h2, aS2, aD2, as2, ad2, N, 1, 40);
  edge_max_k  <<<grid1(Etot), B, 0, stream>>>(ei, as2, ad2, mk2, Eraw, Etot, 1);
  edge_denom_k<<<grid1(Etot), B, 0, stream>>>(ei, as2, ad2, mk2, den2, Eraw, Etot, 1);
  edge_aggr_k <<<grid1(Etot), B, 0, stream>>>(ei, as2, ad2, mk2, den2, h2,
                                              out2, Eraw, Etot, 1, 40);

  final_lsm_k<<<grid1((long long)N), B, 0, stream>>>(out2, b2, (float*)d_out, N);
}